// PoseNMSAndReturnAsBatchedResult_2585570312411
// MI455X (gfx1250) — compile-verified
//
#include <hip/hip_runtime.h>
#include <hip/hip_bf16.h>
#include <stdint.h>

// Problem constants (match reference)
#define BATCH 32
#define NPRE  4096
#define KSEL  300
#define NSEL  (BATCH * KSEL)   // 9600 selection records
#define JN    51               // 17 joints * 3

// 128-bit vector pointee types with explicit address spaces (builtin signature
// verified by compiler diagnostics in earlier rounds).
typedef int v4i __attribute__((ext_vector_type(4)));
typedef __attribute__((address_space(1))) v4i* as1_v4i_p;
typedef __attribute__((address_space(3))) v4i* as3_v4i_p;

// ---------------------------------------------------------------------------
// Single fused kernel: one 1024-thread block (32 wave32s) per batch.
//   1. Async-stage the batch's 4096 scores into LDS (overlapped with key init)
//   2. Scan all 9600 selection records; for records of this batch, count them
//      and splat the score into the 64-bit sort key for that box index
//   3. Bitonic-sort 4096 keys descending in LDS (stable tie-break = low index)
//   4. Gather boxes / scores / poses for the top-300 and write num_predictions
// LDS: 32 KB keys + 16 KB scores + 4 B counter = 48 KB per WGP (of 320 KB).
// ---------------------------------------------------------------------------
__global__ __launch_bounds__(1024) void pose_nms_fused(
    const float* __restrict__ boxes,   // [B, N, 4]
    const float* __restrict__ scores,  // [B, N] (last dim 1 squeezed)
    const float* __restrict__ joints,  // [B, N, 51]
    const int*   __restrict__ sel,     // [NSEL, 3]
    float* __restrict__ out) {
  __shared__ unsigned long long keys[NPRE];   // 32 KB
  __shared__ __align__(16) float ssc[NPRE];   // 16 KB
  __shared__ int scount;

  const int b   = blockIdx.x;
  const int tid = threadIdx.x;
  const float* scB = scores + (size_t)b * NPRE;

  // ---- (1) async DMA of scores into LDS; init keys while it is in flight ----
#if __has_builtin(__builtin_amdgcn_global_load_async_to_lds_b128)
  // each of the 1024 threads moves one 16B chunk; per-lane LDS dest address
  __builtin_amdgcn_global_load_async_to_lds_b128(
      (as1_v4i_p)(scB + tid * 4),
      (as3_v4i_p)(&ssc[tid * 4]),
      /*offset=*/0, /*cpol=*/0);
#else
  ((float4*)ssc)[tid] = ((const float4*)scB)[tid];
#endif

  // default key = masked score 0.0 with inverted index (low index wins ties,
  // matching jax.lax.top_k's stable tie-break under a descending sort)
  for (int i = tid; i < NPRE; i += 1024)
    keys[i] = (unsigned long long)(0xFFFFFFFFu - (unsigned)i);
  if (tid == 0) scount = 0;

#if __has_builtin(__builtin_amdgcn_global_load_async_to_lds_b128)
#if __has_builtin(__builtin_amdgcn_s_wait_asynccnt)
  __builtin_amdgcn_s_wait_asynccnt(0);
#else
  asm volatile("s_wait_asynccnt 0" ::: "memory");
#endif
#endif
  __syncthreads();

  // ---- (2) scan selection records for this batch ----
  // Reference semantics: count every record whose batch id == b (regardless
  // of box validity); mark mask only when box is also in range. Scores are
  // uniform in [0,1) -> non-negative -> float bits are order-monotonic, and
  // duplicate records write identical 64-bit values (benign race).
  for (int t = tid; t < NSEL; t += 1024) {
    int bb  = sel[3 * t + 0];
    int box = sel[3 * t + 2];
    if (bb == b) {
      atomicAdd(&scount, 1);
      if (box >= 0 && box < NPRE) {
        unsigned bits = __float_as_uint(ssc[box]);
        keys[box] = ((unsigned long long)bits << 32) |
                    (unsigned long long)(0xFFFFFFFFu - (unsigned)box);
      }
    }
  }
  __syncthreads();

  // ---- (3) bitonic sort, descending (largest key first) ----
  for (unsigned k = 2; k <= NPRE; k <<= 1) {
    for (unsigned j = k >> 1; j > 0; j >>= 1) {
      for (unsigned i = tid; i < NPRE; i += 1024) {
        unsigned ixj = i ^ j;
        if (ixj > i) {
          unsigned long long a = keys[i];
          unsigned long long c = keys[ixj];
          bool doswap = ((i & k) == 0) ? (a < c) : (a > c);
          if (doswap) { keys[i] = c; keys[ixj] = a; }
        }
      }
      __syncthreads();
    }
  }

  // ---- (4) outputs (flat, in reference return order) ----
  float* out_np     = out;                              // [B]
  float* out_boxes  = out + BATCH;                      // [B, K, 4]
  float* out_scores = out_boxes + BATCH * KSEL * 4;     // [B, K]
  float* out_poses  = out_scores + BATCH * KSEL;        // [B, K, 51]

  const float* bxB = boxes  + (size_t)b * NPRE * 4;
  const float* jtB = joints + (size_t)b * NPRE * JN;

  if (tid == 0) {
    int c = scount;
    out_np[b] = (float)(c < KSEL ? c : KSEL);
  }
  // boxes: 1200 floats per batch
  for (int e = tid; e < KSEL * 4; e += 1024) {
    int kk = e >> 2, c = e & 3;
    unsigned idx = 0xFFFFFFFFu - (unsigned)(keys[kk] & 0xFFFFFFFFull);
    out_boxes[(size_t)b * KSEL * 4 + e] = bxB[(size_t)idx * 4 + c];
  }
  // scores: gather ORIGINAL (unmasked) scores — served from LDS stage
  for (int e = tid; e < KSEL; e += 1024) {
    unsigned idx = 0xFFFFFFFFu - (unsigned)(keys[e] & 0xFFFFFFFFull);
    out_scores[(size_t)b * KSEL + e] = ssc[idx];
  }
  // poses: 15300 floats per batch; each item is 204 contiguous bytes in src
  for (int e = tid; e < KSEL * JN; e += 1024) {
    int kk = e / JN, j = e - kk * JN;
    unsigned idx = 0xFFFFFFFFu - (unsigned)(keys[kk] & 0xFFFFFFFFull);
    out_poses[(size_t)b * KSEL * JN + e] = jtB[(size_t)idx * JN + j];
  }
}

// ---------------------------------------------------------------------------
extern "C" void kernel_launch(void* const* d_in, const int* in_sizes, int n_in,
                              void* d_out, int out_size, void* d_ws, size_t ws_size,
                              hipStream_t stream) {
  const float* boxes  = (const float*)d_in[0];  // [32,4096,4]
  const float* scores = (const float*)d_in[1];  // [32,4096,1]
  const float* joints = (const float*)d_in[2];  // [32,4096,17,3]
  const int*   sel    = (const int*)d_in[3];    // [9600,3]

  // Single launch: one workgroup per batch image.
  pose_nms_fused<<<BATCH, 1024, 0, stream>>>(boxes, scores, joints, sel,
                                             (float*)d_out);
}